// RPN_18124761989760
// MI455X (gfx1250) — compile-verified
//
#include <hip/hip_runtime.h>

typedef __attribute__((ext_vector_type(2))) float v2f;
typedef __attribute__((ext_vector_type(8))) float v8f;

#define T_FRAMES 4096
#define D_DIM    256
#define A_NUM    10
#define KSTEPS   64          // 256 / 4 (K per V_WMMA_F32_16X16X4_F32)
#define NTILES   3           // 48 columns of accumulators, 40 used
#define ROWS_PER_WAVE  32    // 2 WMMA M-tiles per wave
#define ROWS_PER_BLOCK 256   // 8 waves
#define TSTRIDE  50          // LDS tile row stride (conflict padding)

// smem is used twice: first as pre-swizzled B fragments (12288 floats),
// then (after a barrier) as per-wave D spill tiles (8 * 32 * 50 = 12800 floats).
#define SMEM_FLOATS 12800

__global__ __launch_bounds__(256) void rpn_fused_kernel(
    const float* __restrict__ x,
    const float* __restrict__ W_cls,  const float* __restrict__ b_cls,
    const float* __restrict__ W_bbox, const float* __restrict__ b_bbox,
    float* __restrict__ out_prop, float* __restrict__ out_cls,
    float* __restrict__ out_bbox)
{
    __shared__ float smem[SMEM_FLOATS];
    const int tid  = threadIdx.x;
    const int lane = tid & 31;
    const int wave = tid >> 5;

    // ---- Stage combined [W_cls; W_bbox]^T into LDS in WMMA B-fragment layout.
    // Fragment for (kk, nt): lane l (<16): v0=B[4kk+0][nt*16+l], v1=B[4kk+1][..]
    //                        lane l (>=16): v0=B[4kk+2][..], v1=B[4kk+3][..]
    // with B[k][n] = Wcombined[n][k].  Flat: ((kk*3+nt)*32 + lane)*2 + j.
    for (int i = 0; i < 48; ++i) {
        int f   = i * 256 + tid;          // 0..12287
        int j   = f & 1;
        int ln  = (f >> 1) & 31;
        int ntk = f >> 6;                 // kk*3 + nt
        int nt  = ntk % 3;
        int kk  = ntk / 3;
        int n   = nt * 16 + (ln & 15);
        int k   = kk * 4 + ((ln >> 4) << 1) + j;
        float v = 0.0f;
        if (n < 20)      v = W_cls [n * D_DIM + k];
        else if (n < 40) v = W_bbox[(n - 20) * D_DIM + k];
        smem[f] = v;
    }
    __syncthreads();

    const long m0 = (long)blockIdx.x * ROWS_PER_BLOCK + wave * ROWS_PER_WAVE;

    const v8f zero = {0.f, 0.f, 0.f, 0.f, 0.f, 0.f, 0.f, 0.f};
    v8f acc[2][NTILES];
    for (int mt = 0; mt < 2; ++mt)
        for (int nt = 0; nt < NTILES; ++nt)
            acc[mt][nt] = zero;

    // A fragment (16x4 f32): lanes 0-15 -> rows m0..m0+15, K=4kk+{0,1};
    //                        lanes 16-31 -> same rows, K=4kk+{2,3}.
    const float* xA0 = x + (m0 + (lane & 15)) * (long)D_DIM + ((lane >> 4) << 1);
    const float* xA1 = xA0 + 16 * D_DIM;

    #pragma unroll 4
    for (int kk = 0; kk < KSTEPS; ++kk) {
        v2f a0 = *(const v2f*)(xA0 + kk * 4);
        v2f a1 = *(const v2f*)(xA1 + kk * 4);
        const float* fb = smem + kk * (NTILES * 64) + lane * 2;
        v2f b0 = *(const v2f*)(fb);
        v2f b1 = *(const v2f*)(fb + 64);
        v2f b2 = *(const v2f*)(fb + 128);
        acc[0][0] = __builtin_amdgcn_wmma_f32_16x16x4_f32(false, a0, false, b0, (short)0, acc[0][0], false, false);
        acc[0][1] = __builtin_amdgcn_wmma_f32_16x16x4_f32(false, a0, false, b1, (short)0, acc[0][1], false, false);
        acc[0][2] = __builtin_amdgcn_wmma_f32_16x16x4_f32(false, a0, false, b2, (short)0, acc[0][2], false, false);
        acc[1][0] = __builtin_amdgcn_wmma_f32_16x16x4_f32(false, a1, false, b0, (short)0, acc[1][0], false, false);
        acc[1][1] = __builtin_amdgcn_wmma_f32_16x16x4_f32(false, a1, false, b1, (short)0, acc[1][1], false, false);
        acc[1][2] = __builtin_amdgcn_wmma_f32_16x16x4_f32(false, a1, false, b2, (short)0, acc[1][2], false, false);
    }
    __syncthreads();   // everyone done reading B fragments; reuse smem

    // ---- Spill D tiles: VGPR r of lane l holds D[M = r + 8*(l/16)][N = l%16].
    float* tl = smem + wave * (ROWS_PER_WAVE * TSTRIDE);
    const int colB = lane & 15;
    const int rOff = (lane >> 4) << 3;
    #pragma unroll
    for (int mt = 0; mt < 2; ++mt)
        #pragma unroll
        for (int nt = 0; nt < NTILES; ++nt)
            #pragma unroll
            for (int r = 0; r < 8; ++r)
                tl[(mt * 16 + rOff + r) * TSTRIDE + nt * 16 + colB] = acc[mt][nt][r];
    __syncthreads();

    // ---- Epilogue: bias + stores. cls/bbox flat index = m*20 + col, so the
    // wave's 640 stores per head are fully contiguous starting at m0*20.
    #pragma unroll
    for (int i = 0; i < 20; ++i) {
        int jj  = i * 32 + lane;           // 0..639
        int row = jj / 20;
        int col = jj - row * 20;
        float vc = tl[row * TSTRIDE + col]      + b_cls[col];
        float vb = tl[row * TSTRIDE + 20 + col] + b_bbox[col];
        out_cls [m0 * 20 + jj] = vc;
        out_bbox[m0 * 20 + jj] = vb;
    }

    // Proposals: a_ctr = t, a_len = linspace(40,200,10)[a].
    const float stepL = 160.0f / 9.0f;
    #pragma unroll
    for (int i = 0; i < 10; ++i) {
        int jj  = i * 32 + lane;           // 0..319  (32 rows x 10 anchors)
        int row = jj / 10;
        int a   = jj - row * 10;
        long m  = m0 + row;
        float tf  = (float)(m & (T_FRAMES - 1));
        float len = (a == A_NUM - 1) ? 200.0f : fmaf(stepL, (float)a, 40.0f);
        float dx  = tl[row * TSTRIDE + 20 + 2 * a];
        float dl  = tl[row * TSTRIDE + 21 + 2 * a];
        float pc  = dx * len + tf;
        float pl  = expf(dl) * len;
        float h   = 0.5f * (pl - 1.0f);
        float s = fminf(fmaxf(pc - h, 0.0f), (float)(T_FRAMES - 1));
        float e = fminf(fmaxf(pc + h, 0.0f), (float)(T_FRAMES - 1));
        v2f se = {s, e};
        *(v2f*)(out_prop + m * 20 + 2 * a) = se;
    }
}

__global__ __launch_bounds__(256) void anchors_kernel(float* __restrict__ out)
{
    int idx = blockIdx.x * 256 + threadIdx.x;      // T*A = 40960
    if (idx >= T_FRAMES * A_NUM) return;
    int t = idx / A_NUM;
    int a = idx - t * A_NUM;
    const float stepL = 160.0f / 9.0f;
    float len = (a == A_NUM - 1) ? 200.0f : fmaf(stepL, (float)a, 40.0f);
    float h = 0.5f * (len - 1.0f);
    v2f se = {(float)t - h, (float)t + h};
    *(v2f*)(out + idx * 2) = se;
}

extern "C" void kernel_launch(void* const* d_in, const int* in_sizes, int n_in,
                              void* d_out, int out_size, void* d_ws, size_t ws_size,
                              hipStream_t stream) {
    (void)in_sizes; (void)n_in; (void)out_size; (void)d_ws; (void)ws_size;
    const float* x      = (const float*)d_in[0];
    const float* W_cls  = (const float*)d_in[1];
    const float* b_cls  = (const float*)d_in[2];
    const float* W_bbox = (const float*)d_in[3];
    const float* b_bbox = (const float*)d_in[4];

    float* out     = (float*)d_out;
    // Outputs concatenated flat in return order:
    float* anchors = out;                                   // [T*A, 2]
    float* prop    = anchors + (long)T_FRAMES * A_NUM * 2;  // [B, T*A, 2]
    float* cls     = prop    + 64L * T_FRAMES * A_NUM * 2;  // [B, T*A, 2]
    float* bbox    = cls     + 64L * T_FRAMES * A_NUM * 2;  // [B, T*A, 2]

    anchors_kernel<<<(T_FRAMES * A_NUM + 255) / 256, 256, 0, stream>>>(anchors);

    const long rows = 64L * T_FRAMES;                       // B*T = 262144
    rpn_fused_kernel<<<(int)(rows / ROWS_PER_BLOCK), 256, 0, stream>>>(
        x, W_cls, b_cls, W_bbox, b_bbox, prop, cls, bbox);
}